// HybridQuantumClassicalNetwork_73890617361035
// MI455X (gfx1250) — compile-verified
//
#include <hip/hip_runtime.h>

typedef __attribute__((ext_vector_type(2))) float v2f;
typedef __attribute__((ext_vector_type(8))) float v8f;

#define NQ    10        // qubits
#define KDIM  256       // encoder K
#define ENC_N 20        // encoder N (theta|phi)
#define BATCH 8
#define WSP   40        // padded W_enc LDS row: +2 K-rows => +16 banks (conflict-free halves)

// One fused kernel, one workgroup (2 waves). Closed-form quantum layer:
//   p_q(occ1) = (1 + sin(theta_q) cos(phi_q)) / 2   (product state through H + rotation)
//   CX chain  => logical bit k = XOR of bits 0..k
//   q_feats[k] = (1 - prod_{j<=k}(-sin(theta_j) cos(phi_j))) / 2
__global__ __launch_bounds__(64)
void hqcn_fused_kernel(const float* __restrict__ x,      // 8 x 256
                       const float* __restrict__ W_enc,  // 256 x 20
                       const float* __restrict__ b_enc,  // 20
                       const float* __restrict__ W_dec,  // 10 x 10
                       const float* __restrict__ b_dec,  // 10
                       float* __restrict__ out)          // 8 x 10
{
    const int tid  = threadIdx.x;     // 0..63
    const int wv   = tid >> 5;        // wave id: owns encoder N-tile wv
    const int lane = tid & 31;
    const int half = lane >> 4;       // selects K pair {0,1} vs {2,3}
    const int l16  = lane & 15;

    __shared__ float xs_t[KDIM][16];  // x transposed, rows 8..15 zero
    __shared__ float ws[KDIM][WSP];   // W_enc, cols 20..39 zero
    __shared__ float enc_s[16][32];   // encoder C tiles
    __shared__ float m_s[BATCH][NQ];  // m = 1 - 2p = -sin(theta)*cos(phi)
    __shared__ float qf_s[16][12];    // q_feats padded to 16 rows x K=12
    __shared__ float wd_s[12][16];    // W_dec padded to 12 x 16

    // ---- Phase 0: zero all padded LDS operands (once, outside hot loop) ----
    for (int t = tid; t < KDIM * 16;  t += 64) ((float*)xs_t)[t] = 0.0f;
    for (int t = tid; t < KDIM * WSP; t += 64) ((float*)ws)[t]   = 0.0f;
    for (int t = tid; t < 16 * 12;    t += 64) ((float*)qf_s)[t] = 0.0f;
    for (int t = tid; t < 12 * 16;    t += 64) ((float*)wd_s)[t] = 0.0f;
    __syncthreads();

    // ---- Phase 1: fill valid regions (coalesced global reads) --------------
    for (int t = tid; t < BATCH * KDIM; t += 64) {      // x: 8x256
        const int row = t >> 8, col = t & 255;
        xs_t[col][row] = x[t];
    }
    for (int t = tid; t < KDIM * ENC_N; t += 64) {      // W_enc: 256x20
        const int k = t / ENC_N, n = t - k * ENC_N;
        ws[k][n] = W_enc[t];
    }
    for (int t = tid; t < NQ * NQ; t += 64) {           // W_dec: 10x10
        const int k = t / NQ, n = t - k * NQ;
        wd_s[k][n] = W_dec[t];
    }
    __syncthreads();

    // ---- Stage A: enc = x @ W_enc via V_WMMA_F32_16X16X4_F32 ---------------
    // Wave wv computes encoder columns [wv*16, wv*16+16). Branch-free inner loop.
    v8f acc = {};
    const int bn = wv * 16 + l16;                       // this lane's B column
    for (int k = 0; k < KDIM; k += 4) {
        const int ka = k + half * 2;                    // lane-half K pair base
        v2f a;                                          // A: 16x4, lane = M
        a.x = xs_t[ka + 0][l16];
        a.y = xs_t[ka + 1][l16];
        v2f b;                                          // B: 4x16, lane = N
        b.x = ws[ka + 0][bn];
        b.y = ws[ka + 1][bn];
        acc = __builtin_amdgcn_wmma_f32_16x16x4_f32(false, a, false, b,
                                                    (short)0, acc, false, false);
    }
    {   // C layout: VGPR r -> (M = half*8 + r, N = l16), tile column offset wv*16
        const int mbase = half * 8;
        #pragma unroll
        for (int r = 0; r < 8; ++r)
            enc_s[mbase + r][wv * 16 + l16] = acc[r];
    }
    __syncthreads();

    // ---- Stage C: per (batch,qubit) m = -sin(theta)*cos(phi) ---------------
    const float PI_F = 3.14159265358979323846f;
    for (int t = tid; t < BATCH * NQ; t += 64) {
        const int b = t / NQ, q = t - b * NQ;
        const float th = PI_F * tanhf(enc_s[b][q]      + b_enc[q]);
        const float ph = PI_F * tanhf(enc_s[b][NQ + q] + b_enc[NQ + q]);
        m_s[b][q] = -sinf(th) * cosf(ph);
    }
    __syncthreads();

    // ---- Stage D: prefix product -> q_feats --------------------------------
    if (tid < BATCH) {
        float M = 1.0f;
        #pragma unroll
        for (int k = 0; k < NQ; ++k) {
            M *= m_s[tid][k];
            qf_s[tid][k] = (1.0f - M) * 0.5f;           // p1/(p0+p1), sum == 1
        }
    }
    __syncthreads();

    // ---- Stage E: out = q_feats @ W_dec + b_dec via WMMA (wave 0 only) -----
    if (wv == 0) {
        v8f accd = {};
        #pragma unroll
        for (int k = 0; k < 12; k += 4) {
            const int ka = k + half * 2;
            v2f a;
            a.x = qf_s[l16][ka + 0];
            a.y = qf_s[l16][ka + 1];
            v2f b;
            b.x = wd_s[ka + 0][l16];
            b.y = wd_s[ka + 1][l16];
            accd = __builtin_amdgcn_wmma_f32_16x16x4_f32(false, a, false, b,
                                                         (short)0, accd, false, false);
        }
        // rows 0..7 live in lanes 0..15 (VGPR r -> M=r, N=l16)
        if (half == 0 && l16 < NQ) {
            const float bias = b_dec[l16];
            #pragma unroll
            for (int r = 0; r < BATCH; ++r)
                out[r * NQ + l16] = accd[r] + bias;
        }
    }
}

extern "C" void kernel_launch(void* const* d_in, const int* in_sizes, int n_in,
                              void* d_out, int out_size, void* d_ws, size_t ws_size,
                              hipStream_t stream) {
    (void)in_sizes; (void)n_in; (void)out_size; (void)d_ws; (void)ws_size;
    const float* x     = (const float*)d_in[0];
    const float* W_enc = (const float*)d_in[1];
    const float* b_enc = (const float*)d_in[2];
    const float* W_dec = (const float*)d_in[3];
    const float* b_dec = (const float*)d_in[4];
    float* out = (float*)d_out;

    hqcn_fused_kernel<<<1, 64, 0, stream>>>(x, W_enc, b_enc, W_dec, b_dec, out);
}